// ResContextBlock_52158082843100
// MI455X (gfx1250) — compile-verified
//
#include <hip/hip_runtime.h>

typedef __attribute__((ext_vector_type(2))) float v2f;
typedef __attribute__((ext_vector_type(4))) float v4f;
typedef __attribute__((ext_vector_type(8))) float v8f;

#define COUT  32
#define KTAPS 9
#define BN_EPS 1e-5f

// Build the padded feature table (reference's feat_pad) and zero the pad rows:
//   featpad[0..n)   = features, featpad[n] = 0   (16 channels)
//   buf0[n] = 0                                   (32 channels; rows <n written by layers)
__global__ __launch_bounds__(256) void prep_pad(
    const float4* __restrict__ feat4, float4* __restrict__ featpad4,
    float* __restrict__ buf0, int n)
{
    const int i = blockIdx.x * 256 + threadIdx.x;
    const int total4 = n * 4;                 // n*16 floats as float4
    if (i < total4) featpad4[i] = feat4[i];
    if (blockIdx.x == 0) {
        float* featpad = (float*)featpad4;
        if (threadIdx.x < 16) featpad[(size_t)n * 16 + threadIdx.x] = 0.0f;
        if (threadIdx.x < 32) buf0[(size_t)n * 32 + threadIdx.x]    = 0.0f;
    }
}

// One wave computes a 16-point x 32-channel output tile:
//   D[16x32] = sum_k  A_k[16 x CIN] @ W_k[CIN x 32]   (f32 WMMA 16x16x4)
// A fragment layout (ISA 7.12.2, 32-bit A 16x4): lanes 0-15: M=ln, v0=K0,v1=K1;
// lanes 16-31: M=ln, v0=K2,v1=K3.  C/D: VGPR r -> row r (+8 for high half), N=lane&15.
// `feat` has n+1 rows; row n is all zeros (pad), so gathers are unconditional.
template <int CIN, bool ADD_SC>
__global__ __launch_bounds__(256) void subm_conv_bn_relu(
    const float* __restrict__ feat,   // [n+1, CIN], row n == zeros
    const int*   __restrict__ nbr,    // [n, 9], value n == pad row
    const float* __restrict__ W,      // [9, CIN, 32]
    const float* __restrict__ gamma,  // [32]
    const float* __restrict__ beta,   // [32]
    const float* __restrict__ mean,   // [32]
    const float* __restrict__ var,    // [32]
    const float* __restrict__ sc,     // [n, 32] residual (ADD_SC only)
    float*       __restrict__ out,    // [n(+1), 32]
    int n)
{
    // Stage weights TRANSPOSED in LDS once per block: sWt[k][o][c].
    // B fragment {W[k][c][o], W[k][c+1][o]} is then contiguous -> one ds_load_b64.
    __shared__ float sWt[KTAPS * COUT * CIN];
    for (int i = threadIdx.x; i < KTAPS * CIN * COUT; i += 256) {
        const int k = i / (CIN * COUT);
        const int r = i % (CIN * COUT);
        const int c = r / COUT;               // input channel
        const int o = r % COUT;               // output channel
        sWt[(k * COUT + o) * CIN + c] = W[i]; // coalesced global read
    }
    __syncthreads();

    const int lane = threadIdx.x & 31;
    const int half = lane >> 4;      // 0: K0/K1 half, 1: K2/K3 half
    const int ln   = lane & 15;      // A row / B,D column within tile
    const int tile = blockIdx.x * (blockDim.x >> 5) + (threadIdx.x >> 5);
    const int arow = tile * 16 + ln; // point index this lane gathers for
    const int srow = arow < n ? arow : 0;     // clamp -> unconditional loads

    // Hoist all 9 neighbor indices: one clauseable, unconditional load burst.
    int idxs[KTAPS];
#pragma unroll
    for (int k = 0; k < KTAPS; ++k)
        idxs[k] = nbr[srow * KTAPS + k];

    v8f acc0 = {};   // output channels 0..15
    v8f acc1 = {};   // output channels 16..31

    for (int k = 0; k < KTAPS; ++k) {
        const unsigned idx = (unsigned)idxs[k];          // in [0, n]; n == zero row
        const float* src = feat + (size_t)idx * CIN;
        const float* wkt = sWt + k * (COUT * CIN);

#pragma unroll
        for (int kk = 0; kk < CIN / 4; ++kk) {
            const int c = kk * 4 + 2 * half;             // K index of v0 for this lane

            // A fragment: 2 consecutive channels of the gathered (padded) row.
            const v2f a  = *(const v2f*)(src + c);       // 8B-aligned
            // B fragments: contiguous in transposed LDS -> single b64 each.
            const v2f b0 = *(const v2f*)(wkt + ln        * CIN + c);
            const v2f b1 = *(const v2f*)(wkt + (ln + 16) * CIN + c);

            acc0 = __builtin_amdgcn_wmma_f32_16x16x4_f32(
                false, a, false, b0, (short)0, acc0, false, false);
            acc1 = __builtin_amdgcn_wmma_f32_16x16x4_f32(
                false, a, false, b1, (short)0, acc1, false, false);
        }
    }

    // Epilogue: eval-mode BN + ReLU (+ residual), per output channel.
#pragma unroll
    for (int t = 0; t < 2; ++t) {
        const int ch = t * 16 + ln;
        const float scale = gamma[ch] * __frsqrt_rn(var[ch] + BN_EPS);
        const float shift = beta[ch] - mean[ch] * scale;
        const v8f acc = t ? acc1 : acc0;
#pragma unroll
        for (int r = 0; r < 8; ++r) {
            const int orow = tile * 16 + r + 8 * half;
            if (orow < n) {
                float y = fmaxf(acc[r] * scale + shift, 0.0f);
                if (ADD_SC) y += sc[(size_t)orow * COUT + ch];
                out[(size_t)orow * COUT + ch] = y;
            }
        }
    }
}

extern "C" void kernel_launch(void* const* d_in, const int* in_sizes, int n_in,
                              void* d_out, int out_size, void* d_ws, size_t ws_size,
                              hipStream_t stream) {
    (void)n_in; (void)out_size; (void)ws_size;
    const float* features = (const float*)d_in[0];  // [n,16]
    const int*   nbr13    = (const int*)  d_in[1];  // [n,9]
    const int*   nbr31    = (const int*)  d_in[2];  // [n,9]
    const float* w1       = (const float*)d_in[3];  // [9,16,32]
    const float* w12      = (const float*)d_in[4];  // [9,32,32]
    const float* w2       = (const float*)d_in[5];  // [9,16,32]
    const float* w3       = (const float*)d_in[6];  // [9,32,32]
    const float* gamma    = (const float*)d_in[7];  // [4,32]
    const float* beta     = (const float*)d_in[8];
    const float* mean     = (const float*)d_in[9];
    const float* var      = (const float*)d_in[10];

    const int n = in_sizes[0] / 16;
    float* out = (float*)d_out;

    // Workspace layout (all row counts padded where gathered-from):
    float* featpad = (float*)d_ws;                       // [n+1, 16]
    float* buf0    = featpad + (size_t)(n + 1) * 16;     // [n+1, 32] ping (gathered)
    float* buf1    = buf0    + (size_t)(n + 1) * 32;     // [n,   32] shortcut

    // Build padded features + zero pad rows (every call; ws is not persistent).
    {
        const int blocks = (n * 4 + 255) / 256;
        prep_pad<<<blocks, 256, 0, stream>>>(
            (const float4*)features, (float4*)featpad, buf0, n);
    }

    const int tiles  = (n + 15) / 16;
    const int blocks = (tiles + 7) / 8;         // 8 waves (tiles) per 256-thr block
    dim3 grid(blocks), block(256);

    // shortcut branch: conv1x3(w1,nbr13) -> BN0 -> ReLU -> conv3x1(w12,nbr31) -> BN1 -> ReLU
    subm_conv_bn_relu<16, false><<<grid, block, 0, stream>>>(
        featpad, nbr13, w1, gamma + 0, beta + 0, mean + 0, var + 0, nullptr, buf0, n);
    subm_conv_bn_relu<32, false><<<grid, block, 0, stream>>>(
        buf0, nbr31, w12, gamma + 32, beta + 32, mean + 32, var + 32, nullptr, buf1, n);
    // main branch: conv3x1(w2,nbr31) -> BN2 -> ReLU -> conv1x3(w3,nbr13) -> BN3 -> ReLU (+sc)
    subm_conv_bn_relu<16, false><<<grid, block, 0, stream>>>(
        featpad, nbr31, w2, gamma + 64, beta + 64, mean + 64, var + 64, nullptr, buf0, n);
    subm_conv_bn_relu<32, true><<<grid, block, 0, stream>>>(
        buf0, nbr13, w3, gamma + 96, beta + 96, mean + 96, var + 96, buf1, out, n);
}